// GCN_75746043232256
// MI455X (gfx1250) — compile-verified
//
#include <hip/hip_runtime.h>
#include <hip/hip_bf16.h>

typedef float v2f __attribute__((ext_vector_type(2)));
typedef float v8f __attribute__((ext_vector_type(8)));

#define N_IN   256
#define N_HID  256
#define N_CLS  64

__device__ __forceinline__ void atomAddF(float* p, float v) {
    // lowers to global_atomic_add_f32 on gfx1250
    unsafeAtomicAdd(p, v);
}

// ---------------- utility kernels ----------------

__global__ void zero_f4_kernel(float4* __restrict__ p, int n4) {
    int i = blockIdx.x * blockDim.x + threadIdx.x;
    if (i < n4) p[i] = make_float4(0.f, 0.f, 0.f, 0.f);
}

__global__ void degree_kernel(const int* __restrict__ src, const int* __restrict__ dst,
                              float* __restrict__ outdeg, float* __restrict__ indeg,
                              int n_edges) {
    int e = blockIdx.x * blockDim.x + threadIdx.x;
    if (e < n_edges) {
        atomAddF(&outdeg[src[e]], 1.0f);
        atomAddF(&indeg[dst[e]], 1.0f);
    }
}

__global__ void norm_kernel(float* __restrict__ p, int n) {
    int i = blockIdx.x * blockDim.x + threadIdx.x;
    if (i < n) {
        float d = p[i];
        p[i] = rsqrtf(d > 1.0f ? d : 1.0f);
    }
}

// out[row, :] = agg[row, :] * dstnorm[row]   (FEATS = 64)
__global__ void scale_rows_kernel(const float* __restrict__ agg,
                                  const float* __restrict__ dstnorm,
                                  float* __restrict__ out, int total) {
    int i = blockIdx.x * blockDim.x + threadIdx.x;
    if (i < total) {
        int row = i >> 6;
        out[i] = agg[i] * dstnorm[row];
    }
}

// ---------------- edge gather/scatter: agg[dst] += x[src] * (SCALE ? srcnorm[src] : 1) ----------------

template <int FEATS, bool SCALE>
__global__ void edge_scatter_kernel(const float* __restrict__ x,
                                    const int* __restrict__ src,
                                    const int* __restrict__ dst,
                                    const float* __restrict__ srcnorm,
                                    float* __restrict__ agg,
                                    int total_chunks) {
    constexpr int CH = FEATS / 4;  // float4 chunks per edge
    int idx = blockIdx.x * blockDim.x + threadIdx.x;
    if (idx >= total_chunks) return;
    int e = idx / CH;
    int c = idx - e * CH;
    int s = src[e];
    int d = dst[e];
    const float4 v = *(const float4*)(x + (size_t)s * FEATS + c * 4);
    float sc = 1.0f;
    if (SCALE) sc = srcnorm[s];
    float* ap = agg + (size_t)d * FEATS + c * 4;
    atomAddF(ap + 0, v.x * sc);
    atomAddF(ap + 1, v.y * sc);
    atomAddF(ap + 2, v.z * sc);
    atomAddF(ap + 3, v.w * sc);
}

// ---------------- WMMA f32 GEMM:  C = act( diag(rowscale) * A[M x 256] @ W[256 x NCOLS] ) ----------------
// 256 threads = 8 waves. MTILES waves stacked in M (16 rows each), 8/MTILES tiles in N (64 cols each).
// W staged in LDS in K-panels of 64 rows, PAIR-PACKED: lds[p*NCOLS + n] = {W[kp+2p][n], W[kp+2p+1][n]}
// so each B fragment is a single aligned ds_load_b64 into an even VGPR pair (no repacking movs).
// Uses V_WMMA_F32_16X16X4_F32.
// A fragment (16x4 f32): lanes 0-15 hold {A[m][k],A[m][k+1]}, lanes 16-31 hold {A[m][k+2],A[m][k+3]}.
// B fragment (4x16 f32): lanes 0-15 hold {B[k][n],B[k+1][n]},  lanes 16-31 hold {B[k+2][n],B[k+3][n]}.
// C/D (16x16 f32): VGPR r -> (M=r, N=lane) lanes 0-15 ; (M=r+8, N=lane-16) lanes 16-31.

template <int NCOLS, int MTILES, bool RELU>
__global__ void __launch_bounds__(256)
gemm_wmma_kernel(const float* __restrict__ A,
                 const float* __restrict__ rowscale,
                 const float* __restrict__ W,
                 float* __restrict__ C,
                 int M) {
    constexpr int K = 256;
    constexpr int KP = 64;                    // K panel staged in LDS
    constexpr int NTILES = 8 / MTILES;
    constexpr int NWAVE = NCOLS / NTILES;     // columns per wave (64)
    constexpr int NACC = NWAVE / 16;          // 16x16 accum tiles per wave (4)

    __shared__ v2f lds[(KP / 2) * NCOLS];     // pair-packed K-panel of W

    const int tid  = threadIdx.x;
    const int wave = tid >> 5;
    const int lane = tid & 31;
    const int half = lane >> 4;               // 0: lanes 0-15, 1: lanes 16-31
    const int l    = lane & 15;

    const int mtile = wave % MTILES;
    const int ntile = wave / MTILES;

    const int rowBase = blockIdx.x * (MTILES * 16) + mtile * 16;
    int row  = rowBase + l;
    int rowc = row < M ? row : (M - 1);       // branchless clamp: EXEC stays all-1s for WMMA
    const float* __restrict__ Arow = A + (size_t)rowc * K;
    const float sc = rowscale[rowc];

    v8f acc[NACC];
#pragma unroll
    for (int j = 0; j < NACC; ++j) acc[j] = (v8f)0.0f;

    for (int kp = 0; kp < K; kp += KP) {
        __syncthreads();
        // stage + interleave W[kp .. kp+KP) x NCOLS into LDS as row-pairs
        {
            constexpr int QTOT = KP * NCOLS / 8;   // 2-row x 4-col micro-tiles
            constexpr int CQ = NCOLS / 4;
#pragma unroll
            for (int t = 0; t < QTOT / 256; ++t) {
                int q  = t * 256 + tid;
                int p  = q / CQ;                   // pair index (covers rows 2p, 2p+1)
                int c4 = (q - p * CQ) * 4;         // column start
                const float4 r0 = *(const float4*)(W + (size_t)(kp + 2 * p) * NCOLS + c4);
                const float4 r1 = *(const float4*)(W + (size_t)(kp + 2 * p + 1) * NCOLS + c4);
                v2f* dp = lds + p * NCOLS + c4;
                dp[0] = (v2f){r0.x, r1.x};
                dp[1] = (v2f){r0.y, r1.y};
                dp[2] = (v2f){r0.z, r1.z};
                dp[3] = (v2f){r0.w, r1.w};
            }
        }
        __syncthreads();

#pragma unroll
        for (int kk = 0; kk < KP; kk += 4) {
            const int p = (kk >> 1) + half;       // pair row for this half-wave (kb = kk + 2*half)
            v2f a;
            a.x = Arow[kp + kk + 2 * half + 0] * sc;
            a.y = Arow[kp + kk + 2 * half + 1] * sc;
#pragma unroll
            for (int j = 0; j < NACC; ++j) {
                const int n = ntile * NWAVE + j * 16 + l;
                v2f b = lds[p * NCOLS + n];       // single ds_load_b64
                acc[j] = __builtin_amdgcn_wmma_f32_16x16x4_f32(
                    false, a, false, b, (short)0, acc[j], false, false);
            }
        }
    }

    // write back 16x16 tiles
#pragma unroll
    for (int j = 0; j < NACC; ++j) {
        const int n = ntile * NWAVE + j * 16 + l;
#pragma unroll
        for (int r = 0; r < 8; ++r) {
            const int m = rowBase + r + half * 8;
            float v = acc[j][r];
            if (RELU) v = v > 0.0f ? v : 0.0f;
            if (m < M) C[(size_t)m * NCOLS + n] = v;
        }
    }
}

// ---------------- host launch ----------------

extern "C" void kernel_launch(void* const* d_in, const int* in_sizes, int n_in,
                              void* d_out, int out_size, void* d_ws, size_t ws_size,
                              hipStream_t stream) {
    const float* feat = (const float*)d_in[0];   // [n_nodes, 256]
    const int*   src  = (const int*)  d_in[1];   // [n_edges]
    const int*   dst  = (const int*)  d_in[2];   // [n_edges]
    const float* W0   = (const float*)d_in[3];   // [256, 256]
    const float* W1   = (const float*)d_in[4];   // [256, 256]
    const float* W2   = (const float*)d_in[5];   // [256, 64]
    float* out = (float*)d_out;                  // [n_nodes, 64]

    const int n_nodes = in_sizes[0] / N_IN;      // 100000
    const int n_edges = in_sizes[1];             // 3200000

    // workspace layout (floats)
    float* ws = (float*)d_ws;
    float* src_norm = ws;                                  // n_nodes
    float* dst_norm = ws + n_nodes;                        // n_nodes
    float* bufA = ws + 2 * (size_t)n_nodes;                // n_nodes*256
    float* bufB = bufA + (size_t)n_nodes * N_HID;          // n_nodes*256

    const int T = 256;
    const size_t big = (size_t)n_nodes * N_HID;

    // 1) degrees -> norms (in place)
    {
        int n = 2 * n_nodes;
        zero_f4_kernel<<<(n / 4 + T - 1) / T, T, 0, stream>>>((float4*)src_norm, n / 4);
        degree_kernel<<<(n_edges + T - 1) / T, T, 0, stream>>>(src, dst, src_norm, dst_norm, n_edges);
        norm_kernel<<<(n + T - 1) / T, T, 0, stream>>>(src_norm, n);
    }

    const int gridM256 = (n_nodes + 31) / 32;    // MTILES=2 -> 32 rows/block
    const int gridM64  = (n_nodes + 127) / 128;  // MTILES=8 -> 128 rows/block
    const int ch256 = n_edges * (N_HID / 4);
    const int ch64  = n_edges * (N_CLS / 4);

    // 2) Layer 0: agg = scatter(feat * src_norm); h0 = relu((agg * dst_norm) @ W0)
    zero_f4_kernel<<<((int)(big / 4) + T - 1) / T, T, 0, stream>>>((float4*)bufA, (int)(big / 4));
    edge_scatter_kernel<N_HID, true><<<(ch256 + T - 1) / T, T, 0, stream>>>(
        feat, src, dst, src_norm, bufA, ch256);
    gemm_wmma_kernel<N_HID, 2, true><<<gridM256, T, 0, stream>>>(bufA, dst_norm, W0, bufB, n_nodes);

    // 3) Layer 1: agg = scatter(h0 * src_norm); h1 = relu((agg * dst_norm) @ W1)
    zero_f4_kernel<<<((int)(big / 4) + T - 1) / T, T, 0, stream>>>((float4*)bufA, (int)(big / 4));
    edge_scatter_kernel<N_HID, true><<<(ch256 + T - 1) / T, T, 0, stream>>>(
        bufB, src, dst, src_norm, bufA, ch256);
    gemm_wmma_kernel<N_HID, 2, true><<<gridM256, T, 0, stream>>>(bufA, dst_norm, W1, bufB, n_nodes);

    // 4) Layer 2 (matmul-first): xw = (h1 * src_norm) @ W2 ; agg2 = scatter(xw) ; out = agg2 * dst_norm
    gemm_wmma_kernel<N_CLS, 8, false><<<gridM64, T, 0, stream>>>(bufB, src_norm, W2, bufA, n_nodes);
    {
        size_t small = (size_t)n_nodes * N_CLS;
        zero_f4_kernel<<<((int)(small / 4) + T - 1) / T, T, 0, stream>>>((float4*)bufB, (int)(small / 4));
        edge_scatter_kernel<N_CLS, false><<<(ch64 + T - 1) / T, T, 0, stream>>>(
            bufA, src, dst, src_norm, bufB, ch64);
        scale_rows_kernel<<<((int)small + T - 1) / T, T, 0, stream>>>(bufB, dst_norm, out, (int)small);
    }
}